// AdaptiveMatrixContainer_73349451481729
// MI455X (gfx1250) — compile-verified
//
#include <hip/hip_runtime.h>

// CDNA5 (gfx1250) fused  softmax(relu(A @ B))  for A:[8192,10], B:[10,8192].
// Write-once design: score tiles are computed twice via V_WMMA_F32_16X16X4_F32
// (recompute is ~free at K=10), output hits HBM exactly once (256 MB floor).
//
// K=10 is padded to 12 (3 WMMA K-blocks of 4). Zero-padding lives ONLY in the
// A register operand; B row indices are clamped so B loads are unconditional
// (0 * finite == 0), keeping the inner loop free of EXEC-mask branches.

typedef float v2f __attribute__((ext_vector_type(2)));
typedef float v8f __attribute__((ext_vector_type(8)));

#define N_NODES 8192
#define RANKK   10
#define TILE    16
#define NWAVES  8
#define THREADS (NWAVES * 32)
#define NTILES  (N_NODES / TILE)   // 512 column tiles of 16

__global__ __launch_bounds__(THREADS)
void fused_gemm_relu_softmax(const float* __restrict__ A,   // [8192, 10]
                             const float* __restrict__ B,   // [10, 8192]
                             float* __restrict__ out)       // [8192, 8192]
{
    const int wave = threadIdx.x >> 5;
    const int lane = threadIdx.x & 31;
    const int nloc = lane & 15;       // column within 16-wide tile
    const int hi   = lane >> 4;       // 0: lanes 0-15, 1: lanes 16-31
    const int row0 = blockIdx.x * TILE;

    // ---- A tile in WMMA 16x4 f32 layout, K padded 10 -> 12 (3 K-blocks) ----
    // lanes 0-15: (K=kb*4+0, kb*4+1) ; lanes 16-31: (K=kb*4+2, kb*4+3); M = lane&15
    v2f a[3];
    int kc[3];                        // clamped B row index per K-block (loop-invariant)
    {
        const int m = row0 + nloc;
        #pragma unroll
        for (int kb = 0; kb < 3; ++kb) {
            const int k = kb * 4 + 2 * hi;
            a[kb].x = (k     < RANKK) ? A[m * RANKK + k]     : 0.0f;
            a[kb].y = (k + 1 < RANKK) ? A[m * RANKK + k + 1] : 0.0f;
            // clamp so loads stay in-bounds; padded slots are killed by a==0
            kc[kb]  = (k + 1 < RANKK) ? k : (RANKK - 2);
        }
    }

    __shared__ float red[NWAVES][TILE];
    __shared__ float inv_rowsum[TILE];

    // slot r holds partials for output row (row0 + r + 8*hi)
    float rsum[8];
    #pragma unroll
    for (int r = 0; r < 8; ++r) rsum[r] = 0.0f;

    // ---------------- pass 1: per-row sum of exp(relu(score)) ----------------
    #pragma unroll 2
    for (int t = wave; t < NTILES; t += NWAVES) {
        const float* __restrict__ Bp = B + t * TILE + nloc;
        v8f c = {};
        #pragma unroll
        for (int kb = 0; kb < 3; ++kb) {
            v2f b;
            b.x = Bp[(size_t) kc[kb]      * N_NODES];
            b.y = Bp[(size_t)(kc[kb] + 1) * N_NODES];
            c = __builtin_amdgcn_wmma_f32_16x16x4_f32(
                    false, a[kb], false, b, (short)0, c, false, false);
        }
        #pragma unroll
        for (int r = 0; r < 8; ++r) {
            float s = fminf(fmaxf(c[r], 0.0f), 70.0f);  // relu + overflow guard
            rsum[r] += __expf(s);
        }
    }

    // reduce across the 16 lanes of each half-wave (rows stay within a half)
    #pragma unroll
    for (int r = 0; r < 8; ++r) {
        float v = rsum[r];
        #pragma unroll
        for (int m = 1; m <= 8; m <<= 1) v += __shfl_xor(v, m, 32);
        rsum[r] = v;
    }
    if (nloc == 0) {
        #pragma unroll
        for (int r = 0; r < 8; ++r) red[wave][hi * 8 + r] = rsum[r];
    }
    __syncthreads();
    if (threadIdx.x < TILE) {
        float tot = 0.0f;
        #pragma unroll
        for (int w = 0; w < NWAVES; ++w) tot += red[w][threadIdx.x];
        inv_rowsum[threadIdx.x] = 1.0f / tot;
    }
    __syncthreads();

    float inv[8];
    #pragma unroll
    for (int r = 0; r < 8; ++r) inv[r] = inv_rowsum[hi * 8 + r];

    // ---------------- pass 2: recompute and stream the normalized output ----
    #pragma unroll 2
    for (int t = wave; t < NTILES; t += NWAVES) {
        const int col = t * TILE + nloc;
        const float* __restrict__ Bp = B + col;
        v8f c = {};
        #pragma unroll
        for (int kb = 0; kb < 3; ++kb) {
            v2f b;
            b.x = Bp[(size_t) kc[kb]      * N_NODES];
            b.y = Bp[(size_t)(kc[kb] + 1) * N_NODES];
            c = __builtin_amdgcn_wmma_f32_16x16x4_f32(
                    false, a[kb], false, b, (short)0, c, false, false);
        }
        #pragma unroll
        for (int r = 0; r < 8; ++r) {
            float s = fminf(fmaxf(c[r], 0.0f), 70.0f);
            const size_t idx = (size_t)(row0 + hi * 8 + r) * N_NODES + col;
            __builtin_nontemporal_store(__expf(s) * inv[r], out + idx);
        }
    }
}

extern "C" void kernel_launch(void* const* d_in, const int* in_sizes, int n_in,
                              void* d_out, int out_size, void* d_ws, size_t ws_size,
                              hipStream_t stream) {
    const float* nv1 = (const float*)d_in[0];   // [8192, 10]
    const float* nv2 = (const float*)d_in[1];   // [10, 8192]
    float* out = (float*)d_out;                 // [8192, 8192]

    dim3 grid(N_NODES / TILE);   // 512 blocks, one 16-row tile each
    dim3 block(THREADS);         // 8 waves of 32
    hipLaunchKernelGGL(fused_gemm_relu_softmax, grid, block, 0, stream,
                       nv1, nv2, out);
}